// HKANGNN_66597762892080
// MI455X (gfx1250) — compile-verified
//
#include <hip/hip_runtime.h>
#include <math.h>

// ---------------------------------------------------------------------------
// HKANGNN fused forward for MI455X (gfx1250, wave32, WMMA f32 16x16x4,
// async global->LDS double-buffered staging).
//
// out[n, o] = KAN(relu(out_e)), out_e collapsed to one K=784 GEMM:
//   out_e = x_email @ Wc^T + extra @ Wx^T
// where Wc = (w_root_se + w_root_ue) @ w_email  (128x768)
//       extra[n, 0:16] = [ssum, deg_s, xsum[0:8], deg_u, 1, 0,0,0,0]
//       Wx columns     = [v1,   v2,    Wu[:,0:8], vu,  bias_c, 0..]
// (rank-1 sender branch, rank-8 url branch folded into the GEMM).
// ---------------------------------------------------------------------------

#define HID   128
#define NE    100000
#define NS    30000
#define NU    50000
#define E_ES  100000
#define E_EU  300000
#define KX    768
#define KE    16
#define KTOT  784      // 768 + 16
#define NCHUNK 49      // 784 / 16
#define LDSP  20       // padded row stride (floats): 16B-aligned + conflict-free

#define USE_ASYNC_LDS 1   // gfx1250 GLOBAL_LOAD_ASYNC_TO_LDS_B128 staging

typedef __attribute__((ext_vector_type(2))) float v2f;
typedef __attribute__((ext_vector_type(8))) float v8f;

__device__ __forceinline__ float gknot(int j) { return 0.4f * (float)j - 2.2f; }

#if USE_ASYNC_LDS
// One 16-byte async copy global->LDS per lane. lds_off = low 32 bits of the
// generic pointer to shared memory (flat aperture: addr[31:0] is the LDS
// byte offset).  Tracked by ASYNCcnt; completes in order within a wave.
__device__ __forceinline__ void async_copy16(const float* gaddr, float* laddr)
{
    unsigned lds_off = (unsigned)(unsigned long long)laddr;
    asm volatile("global_load_async_to_lds_b128 %0, %1, off"
                 :: "v"(lds_off), "v"(gaddr) : "memory");
}
__device__ __forceinline__ void async_wait4()
{
    asm volatile("s_wait_asynccnt 0x4" ::: "memory");
}
__device__ __forceinline__ void async_wait0()
{
    asm volatile("s_wait_asynccnt 0x0" ::: "memory");
}
#endif

// ---------------------------------------------------------------------------
// Build fused weight matrix W[128, 784] (row-major) on device.
// ---------------------------------------------------------------------------
__global__ void prep_w_kernel(
    const float* __restrict__ w_email,  const float* __restrict__ b_email,
    const float* __restrict__ w_url,    const float* __restrict__ b_url,
    const float* __restrict__ w_sender, const float* __restrict__ b_sender,
    const float* __restrict__ w_rel_se, const float* __restrict__ b_rel_se,
    const float* __restrict__ w_root_se,
    const float* __restrict__ w_rel_ue, const float* __restrict__ b_rel_ue,
    const float* __restrict__ w_root_ue,
    float* __restrict__ W)
{
    int idx = blockIdx.x * blockDim.x + threadIdx.x;
    if (idx >= HID * KTOT) return;
    int o = idx / KTOT;
    int k = idx - o * KTOT;
    float v = 0.0f;
    if (k < KX) {                      // combined root path
        for (int h = 0; h < HID; ++h)
            v += (w_root_se[o * HID + h] + w_root_ue[o * HID + h]) * w_email[h * KX + k];
    } else if (k == KX + 0) {          // v1: w_rel_se @ w_sender
        for (int h = 0; h < HID; ++h) v += w_rel_se[o * HID + h] * w_sender[h];
    } else if (k == KX + 1) {          // v2: w_rel_se @ b_sender
        for (int h = 0; h < HID; ++h) v += w_rel_se[o * HID + h] * b_sender[h];
    } else if (k >= KX + 2 && k < KX + 10) {  // Wu[:,kk] = w_rel_ue @ w_url[:,kk]
        int kk = k - (KX + 2);
        for (int h = 0; h < HID; ++h) v += w_rel_ue[o * HID + h] * w_url[h * 8 + kk];
    } else if (k == KX + 10) {         // vu: w_rel_ue @ b_url
        for (int h = 0; h < HID; ++h) v += w_rel_ue[o * HID + h] * b_url[h];
    } else if (k == KX + 11) {         // bias_c
        v = b_rel_se[o] + b_rel_ue[o];
        for (int h = 0; h < HID; ++h)
            v += (w_root_se[o * HID + h] + w_root_ue[o * HID + h]) * b_email[h];
    }                                  // k = KX+12..15 stay zero
    W[o * KTOT + k] = v;
}

// extra[NE,16] init: zeros except col 11 = 1.0 (constant bias feature)
__global__ void init_extra_kernel(float* __restrict__ extra)
{
    int idx = blockIdx.x * blockDim.x + threadIdx.x;
    if (idx >= NE * KE) return;
    extra[idx] = ((idx & 15) == 11) ? 1.0f : 0.0f;
}

// sender->email edges: scalar scatter (ssum to col0, degree to col1)
__global__ void scatter_se_kernel(const int* __restrict__ src,
                                  const int* __restrict__ dst,
                                  const float* __restrict__ x_sender,
                                  float* __restrict__ extra)
{
    int i = blockIdx.x * blockDim.x + threadIdx.x;
    if (i >= E_ES) return;
    int s = src[i], d = dst[i];
    atomicAdd(&extra[d * KE + 0], x_sender[s]);
    atomicAdd(&extra[d * KE + 1], 1.0f);
}

// url->email edges: 8-float scatter (xsum to cols 2..9, degree to col10)
__global__ void scatter_ue_kernel(const int* __restrict__ src,
                                  const int* __restrict__ dst,
                                  const float* __restrict__ x_url,
                                  float* __restrict__ extra)
{
    int gid = blockIdx.x * blockDim.x + threadIdx.x;
    if (gid >= E_EU * 8) return;
    int e = gid >> 3;
    int k = gid & 7;
    int s = src[e], d = dst[e];
    atomicAdd(&extra[d * KE + 2 + k], x_url[s * 8 + k]);
    if (k == 0) atomicAdd(&extra[d * KE + 10], 1.0f);
}

// ---------------------------------------------------------------------------
// Stage one 16-wide K chunk (A: 128 rows, B: 128 weight rows) into LDS.
// Every thread issues exactly 4 x 16B transfers (uniform -> exact ASYNCcnt).
// OOB rows are clamped: their garbage only affects never-written D rows.
// ---------------------------------------------------------------------------
__device__ __forceinline__ void stage_chunk(
    int c, float* dstA, float* dstB,
    const float* __restrict__ x_email, const float* __restrict__ extra,
    const float* __restrict__ W, int m0, int tid)
{
    const bool mainPhase = (c < 48);
    const int kbase = mainPhase ? c * 16 : 0;
    #pragma unroll
    for (int it = 0; it < 2; ++it) {
        int idx4 = tid + it * 256;                // 0..511 float4 slots
        int r  = idx4 >> 2;                       // 0..127
        int k4 = (idx4 & 3) * 4;                  // 0,4,8,12
        int grow = m0 + r;
        if (grow > NE - 1) grow = NE - 1;         // clamp (see note above)
        const float* ga = mainPhase ? (x_email + (size_t)grow * KX + kbase + k4)
                                    : (extra + (size_t)grow * KE + k4);
        const float* gb = W + r * KTOT + (mainPhase ? (kbase + k4) : (KX + k4));
        float* la = dstA + r * LDSP + k4;
        float* lb = dstB + r * LDSP + k4;
#if USE_ASYNC_LDS
        async_copy16(ga, la);
        async_copy16(gb, lb);
#else
        *(float4*)la = *(const float4*)ga;
        *(float4*)lb = *(const float4*)gb;
#endif
    }
}

// ---------------------------------------------------------------------------
// Fused GEMM (K=784, f32 WMMA) + ReLU + KAN (SiLU base + cubic B-splines)
// Block: 256 threads = 8 waves; tile 128 rows x 128 cols; double-buffered LDS.
// ---------------------------------------------------------------------------
__global__ __launch_bounds__(256) void fused_gemm_kan_kernel(
    const float* __restrict__ x_email,
    const float* __restrict__ extra,
    const float* __restrict__ W,
    const float* __restrict__ base_w,
    const float* __restrict__ spline_w,
    float* __restrict__ out)
{
    __shared__ float lA[2][128 * LDSP];   // A tiles (double buffered)
    __shared__ float lB[2][128 * LDSP];   // B tiles (double buffered)
    __shared__ float lP[8 * 128 * 2];     // per-wave partial KAN sums

    const int tid  = threadIdx.x;
    const int wv   = tid >> 5;            // wave 0..7 (wave32)
    const int lane = tid & 31;
    const int half = lane >> 4;
    const int l16  = lane & 15;
    const int m0   = blockIdx.x * 128;

    v8f acc[8];
    {
        v8f z = {0.f, 0.f, 0.f, 0.f, 0.f, 0.f, 0.f, 0.f};
        #pragma unroll
        for (int mt = 0; mt < 8; ++mt) acc[mt] = z;
    }

    // prologue: start chunk 0
    stage_chunk(0, lA[0], lB[0], x_email, extra, W, m0, tid);

    for (int c = 0; c < NCHUNK; ++c) {
        // kick off chunk c+1 into the other buffer, then drain chunk c
        if (c + 1 < NCHUNK) {
            stage_chunk(c + 1, lA[(c + 1) & 1], lB[(c + 1) & 1],
                        x_email, extra, W, m0, tid);
#if USE_ASYNC_LDS
            async_wait4();        // 4 newest (c+1) may remain; chunk c done
        } else {
            async_wait0();
#endif
        }
        __syncthreads();

        const float* As = lA[c & 1];
        const float* Bs = lB[c & 1];
        #pragma unroll
        for (int kk = 0; kk < 16; kk += 4) {
            const int bk = kk + half * 2;           // lanes 16-31: K+2,K+3
            v2f bf = *(const v2f*)&Bs[(wv * 16 + l16) * LDSP + bk];
            v2f af[8];
            #pragma unroll
            for (int mt = 0; mt < 8; ++mt)          // batch loads...
                af[mt] = *(const v2f*)&As[(mt * 16 + l16) * LDSP + bk];
            #pragma unroll
            for (int mt = 0; mt < 8; ++mt)          // ...then batch WMMAs
                acc[mt] = __builtin_amdgcn_wmma_f32_16x16x4_f32(
                    false, af[mt], false, bf, (short)0, acc[mt], false, false);
        }
        __syncthreads();   // buffer c&1 free for reuse at iteration c+1's stage
    }

    // ---------------- fused KAN epilogue ----------------
    const int col = wv * 16 + l16;                  // channel for this lane
    const float bw0 = base_w[col];
    const float bw1 = base_w[HID + col];
    float sw0[8], sw1[8];
    #pragma unroll
    for (int j = 0; j < 8; ++j) {
        sw0[j] = spline_w[col * 8 + j];
        sw1[j] = spline_w[(HID + col) * 8 + j];
    }

    #pragma unroll
    for (int mt = 0; mt < 8; ++mt) {
        #pragma unroll
        for (int v = 0; v < 8; ++v) {
            float h = acc[mt][v];
            h = fmaxf(h, 0.0f);                     // relu (bias folded in)
            float sig = 1.0f / (1.0f + __expf(-h));
            float c0 = h * sig * bw0;               // SiLU base branch
            float c1 = h * sig * bw1;
            // Cox-de Boor: 12 uniform knots, degree 3 -> 8 bases
            float bs[11];
            #pragma unroll
            for (int j = 0; j < 11; ++j)
                bs[j] = (h >= gknot(j) && h < gknot(j + 1)) ? 1.0f : 0.0f;
            #pragma unroll
            for (int p = 1; p <= 3; ++p) {
                const float inv = 1.0f / (0.4f * (float)p);
                #pragma unroll
                for (int j = 0; j < 11 - p; ++j)
                    bs[j] = (h - gknot(j)) * inv * bs[j]
                          + (gknot(j + p + 1) - h) * inv * bs[j + 1];
            }
            #pragma unroll
            for (int j = 0; j < 8; ++j) {
                c0 += bs[j] * sw0[j];
                c1 += bs[j] * sw1[j];
            }
            // reduce across the 16 lanes of each half
            #pragma unroll
            for (int off = 8; off >= 1; off >>= 1) {
                c0 += __shfl_xor(c0, off, 32);
                c1 += __shfl_xor(c1, off, 32);
            }
            if (l16 == 0) {
                int lr = mt * 16 + v + half * 8;    // local row 0..127
                lP[(wv * 128 + lr) * 2 + 0] = c0;
                lP[(wv * 128 + lr) * 2 + 1] = c1;
            }
        }
    }
    __syncthreads();
    // cross-wave reduction: 256 threads -> 128 rows x 2 outputs
    {
        int lr = tid >> 1;
        int o  = tid & 1;
        float s = 0.0f;
        #pragma unroll
        for (int w2 = 0; w2 < 8; ++w2) s += lP[(w2 * 128 + lr) * 2 + o];
        int grow = m0 + lr;
        if (grow < NE) out[grow * 2 + o] = s;
    }
}

// ---------------------------------------------------------------------------
extern "C" void kernel_launch(void* const* d_in, const int* in_sizes, int n_in,
                              void* d_out, int out_size, void* d_ws, size_t ws_size,
                              hipStream_t stream)
{
    const float* x_email   = (const float*)d_in[0];
    const float* x_url     = (const float*)d_in[1];
    const float* x_sender  = (const float*)d_in[2];
    const float* w_email   = (const float*)d_in[3];
    const float* b_email   = (const float*)d_in[4];
    const float* w_url     = (const float*)d_in[5];
    const float* b_url     = (const float*)d_in[6];
    const float* w_sender  = (const float*)d_in[7];
    const float* b_sender  = (const float*)d_in[8];
    // 9..14: es / eu relation params (dead code in reference output)
    const float* w_rel_se  = (const float*)d_in[15];
    const float* b_rel_se  = (const float*)d_in[16];
    const float* w_root_se = (const float*)d_in[17];
    const float* w_rel_ue  = (const float*)d_in[18];
    const float* b_rel_ue  = (const float*)d_in[19];
    const float* w_root_ue = (const float*)d_in[20];
    const float* base_w    = (const float*)d_in[21];
    const float* spline_w  = (const float*)d_in[22];
    // 23..26: es/eu edges (dead), 27..30: se/ue edges (live)
    const int* se_src = (const int*)d_in[27];
    const int* se_dst = (const int*)d_in[28];
    const int* ue_src = (const int*)d_in[29];
    const int* ue_dst = (const int*)d_in[30];

    float* W     = (float*)d_ws;           // 128*784 floats
    float* extra = W + HID * KTOT;         // NE*16 floats

    prep_w_kernel<<<(HID * KTOT + 255) / 256, 256, 0, stream>>>(
        w_email, b_email, w_url, b_url, w_sender, b_sender,
        w_rel_se, b_rel_se, w_root_se, w_rel_ue, b_rel_ue, w_root_ue, W);

    init_extra_kernel<<<(NE * KE + 255) / 256, 256, 0, stream>>>(extra);

    scatter_se_kernel<<<(E_ES + 255) / 256, 256, 0, stream>>>(
        se_src, se_dst, x_sender, extra);

    scatter_ue_kernel<<<(E_EU * 8 + 255) / 256, 256, 0, stream>>>(
        ue_src, ue_dst, x_url, extra);

    const int mtiles = NE / 16;                       // 6250
    const int blocks = (mtiles + 7) / 8;              // 782
    fused_gemm_kan_kernel<<<blocks, 256, 0, stream>>>(
        x_email, extra, W, base_w, spline_w, (float*)d_out);
}